// GCN_44452911513781
// MI455X (gfx1250) — compile-verified
//
#include <hip/hip_runtime.h>

typedef __attribute__((ext_vector_type(2))) float v2f;
typedef __attribute__((ext_vector_type(8))) float v8f;

#define EPS 1e-5f

// ---------------------------------------------------------------------------
// Degree / normalization kernels
// ---------------------------------------------------------------------------
__global__ void deg_init_k(float* __restrict__ deg, int n) {
    int i = blockIdx.x * blockDim.x + threadIdx.x;
    if (i < n) deg[i] = 1.0f;  // self-loop contributes 1
}

__global__ void deg_count_k(const int* __restrict__ col, float* __restrict__ deg, int nE) {
    int e = blockIdx.x * blockDim.x + threadIdx.x;
    if (e < nE) atomicAdd(&deg[col[e]], 1.0f);
}

__global__ void deg_rsqrt_k(float* __restrict__ deg, int n) {
    int i = blockIdx.x * blockDim.x + threadIdx.x;
    if (i < n) deg[i] = rsqrtf(deg[i]);  // deg >= 1 always
}

__global__ void edge_norm_k(const int* __restrict__ row, const int* __restrict__ col,
                            const float* __restrict__ dinv, float* __restrict__ norm, int nE) {
    int e = blockIdx.x * blockDim.x + threadIdx.x;
    if (e < nE) norm[e] = dinv[row[e]] * dinv[col[e]];
}

// ---------------------------------------------------------------------------
// FP32 WMMA GEMM: C[M x FOUT] = A[M x K] * B[K x FOUT], K/FOUT compile-time.
// One wave (32 threads) per block computes a 16 x (16*NT) tile of C.
// V_WMMA_F32_16X16X4_F32 layouts (wave32):
//   A 16x4 : lanes 0-15 -> {K=k0,k0+1}, lanes 16-31 -> {K=k0+2,k0+3}, M = lane%16
//   B 4x16 : lanes 0-15 -> {K=k0,k0+1}, lanes 16-31 -> {K=k0+2,k0+3}, N = lane%16
//   C 16x16: VGPR j -> row j (lanes 0-15) / row j+8 (lanes 16-31), N = lane%16
// ---------------------------------------------------------------------------
template <int NT, int K, int FOUT>
__global__ __launch_bounds__(32) void gemm_wmma_f32_k(const float* __restrict__ A,
                                                      const float* __restrict__ B,
                                                      float* __restrict__ C) {
    const int colBase = blockIdx.x * (16 * NT);
    const int rowBase = blockIdx.y * 16;
    const int lane = threadIdx.x;
    const int half = lane >> 4;   // 0: K pair {0,1}, 1: K pair {2,3}
    const int lr   = lane & 15;

    v8f acc[NT];
    v8f zero = {};
#pragma unroll
    for (int t = 0; t < NT; ++t) acc[t] = zero;

    const float* Arow = A + (unsigned)(rowBase + lr) * K;

#pragma unroll 4
    for (int k0 = 0; k0 < K; k0 += 4) {
        const int ka = k0 + 2 * half;
        v2f a;
        a.x = Arow[ka];
        a.y = Arow[ka + 1];
#pragma unroll
        for (int t = 0; t < NT; ++t) {
            const float* Bp = B + (unsigned)ka * FOUT + colBase + 16 * t + lr;
            v2f b;
            b.x = Bp[0];
            b.y = Bp[FOUT];
            acc[t] = __builtin_amdgcn_wmma_f32_16x16x4_f32(
                /*neg_a=*/false, a, /*neg_b=*/false, b,
                /*c_mod=*/(short)0, acc[t], /*reuse_a=*/false, /*reuse_b=*/false);
        }
    }

#pragma unroll
    for (int t = 0; t < NT; ++t) {
        float* Cp = C + (unsigned)(rowBase + 8 * half) * FOUT + colBase + 16 * t + lr;
#pragma unroll
        for (int j = 0; j < 8; ++j) Cp[(unsigned)j * FOUT] = acc[t][j];
    }
}

// ---------------------------------------------------------------------------
// Aggregation: out[col] += h[row] * dinv[row] * dinv[col], self-loops folded
// into the init pass (out = h * dinv^2). float4-vectorized; F4 = F/4 is a
// compile-time power of two so index math is shift/mask.
// ---------------------------------------------------------------------------
template <int F4>
__global__ void agg_init_k(const float* __restrict__ h, const float* __restrict__ dinv,
                           float* __restrict__ out, unsigned total /* = N*F4 */) {
    unsigned tid = blockIdx.x * blockDim.x + threadIdx.x;
    if (tid >= total) return;
    unsigned i = tid / F4;      // compile-time const -> shift
    unsigned c = tid % F4;
    float dv = dinv[i];
    float w = dv * dv;
    float4 hv = ((const float4*)h)[(size_t)i * F4 + c];
    float4 ov;
    ov.x = hv.x * w; ov.y = hv.y * w; ov.z = hv.z * w; ov.w = hv.w * w;
    ((float4*)out)[(size_t)i * F4 + c] = ov;
}

template <int F4>
__global__ void scatter_edges_k(const float* __restrict__ h, const int* __restrict__ row,
                                const int* __restrict__ col, const float* __restrict__ norm,
                                float* __restrict__ out, unsigned total /* = E*F4 */) {
    unsigned tid = blockIdx.x * blockDim.x + threadIdx.x;
    if (tid >= total) return;
    unsigned e = tid / F4;      // shift
    unsigned c = tid % F4;      // mask
    if (F4 == 32) {
        // e is wave-uniform (wave32, F4 == 32): force scalar so row/col/norm
        // become scalar loads broadcast to the wave instead of 32 vector loads.
        e = (unsigned)__builtin_amdgcn_readfirstlane((int)e);
    }
    float w = norm[e];
    int r = row[e];
    int d = col[e];
    float4 hv = ((const float4*)h)[(size_t)r * F4 + c];
    float* o = out + ((size_t)d * F4 + c) * 4;
    atomicAdd(o + 0, hv.x * w);
    atomicAdd(o + 1, hv.y * w);
    atomicAdd(o + 2, hv.z * w);
    atomicAdd(o + 3, hv.w * w);
}

// ---------------------------------------------------------------------------
// BN affine precompute:  sc = g * rsqrt(v + eps),  sh = be + (b - m) * sc
// Then BN(x) = x * sc + sh, and ReLU on top.
// ---------------------------------------------------------------------------
__global__ void bn_prep_k(const float* __restrict__ b, const float* __restrict__ g,
                          const float* __restrict__ be, const float* __restrict__ m,
                          const float* __restrict__ v, float* __restrict__ sc,
                          float* __restrict__ sh, int F) {
    int f = blockIdx.x * blockDim.x + threadIdx.x;
    if (f >= F) return;
    float s = g[f] * rsqrtf(v[f] + EPS);
    sc[f] = s;
    sh[f] = be[f] + (b[f] - m[f]) * s;
}

template <int F>
__global__ void bn_relu_k(float* __restrict__ x, const float* __restrict__ sc,
                          const float* __restrict__ sh, unsigned total /* = N*F */) {
    unsigned tid = blockIdx.x * blockDim.x + threadIdx.x;
    if (tid >= total) return;
    unsigned f = tid % F;       // mask
    float y = fmaf(x[tid], sc[f], sh[f]);
    x[tid] = y > 0.0f ? y : 0.0f;
}

// ---------------------------------------------------------------------------
// Final: BN2+ReLU then Linear 64 -> 2. One thread per node.
// ---------------------------------------------------------------------------
__global__ void bn_relu_fc_k(const float* __restrict__ x, const float* __restrict__ sc,
                             const float* __restrict__ sh, const float* __restrict__ fcW,
                             const float* __restrict__ fcb, float* __restrict__ out,
                             int nNodes) {
    int i = blockIdx.x * blockDim.x + threadIdx.x;
    if (i >= nNodes) return;
    const float* xi = x + (size_t)i * 64;
    float o0 = fcb[0], o1 = fcb[1];
#pragma unroll 8
    for (int k = 0; k < 64; ++k) {
        float y = fmaf(xi[k], sc[k], sh[k]);
        y = y > 0.0f ? y : 0.0f;
        o0 = fmaf(y, fcW[k * 2 + 0], o0);
        o1 = fmaf(y, fcW[k * 2 + 1], o1);
    }
    out[(size_t)i * 2 + 0] = o0;
    out[(size_t)i * 2 + 1] = o1;
}

// ---------------------------------------------------------------------------
// Launch
// ---------------------------------------------------------------------------
static inline size_t alignf(size_t n) { return (n + 127) & ~(size_t)127; }

extern "C" void kernel_launch(void* const* d_in, const int* in_sizes, int n_in,
                              void* d_out, int out_size, void* d_ws, size_t ws_size,
                              hipStream_t stream) {
    const float* x    = (const float*)d_in[0];
    const int*   ei   = (const int*)d_in[1];
    const float* W1   = (const float*)d_in[2];
    const float* b1   = (const float*)d_in[3];
    const float* g1   = (const float*)d_in[4];
    const float* be1  = (const float*)d_in[5];
    const float* m1   = (const float*)d_in[6];
    const float* v1   = (const float*)d_in[7];
    const float* W2   = (const float*)d_in[8];
    const float* b2   = (const float*)d_in[9];
    const float* g2   = (const float*)d_in[10];
    const float* be2  = (const float*)d_in[11];
    const float* m2   = (const float*)d_in[12];
    const float* v2   = (const float*)d_in[13];
    const float* fcW  = (const float*)d_in[14];
    const float* fcb  = (const float*)d_in[15];
    float* out = (float*)d_out;

    const int N  = in_sizes[0] / 256;   // 100000 nodes
    const int E  = in_sizes[1] / 2;     // 1600000 edges
    const int* row = ei;
    const int* col = ei + E;

    // workspace layout (floats)
    float* ws = (float*)d_ws;
    size_t off = 0;
    float* dinv = ws + off; off += alignf((size_t)N);
    float* norm = ws + off; off += alignf((size_t)E);
    float* h1   = ws + off; off += alignf((size_t)N * 128);   // reused for h2
    float* agg1 = ws + off; off += alignf((size_t)N * 128);
    float* agg2 = ws + off; off += alignf((size_t)N * 64);
    float* sc1  = ws + off; off += alignf(128);
    float* sh1  = ws + off; off += alignf(128);
    float* sc2  = ws + off; off += alignf(64);
    float* sh2  = ws + off; off += alignf(64);
    float* h2   = h1;
    (void)ws_size; (void)n_in; (void)out_size;

    const int TB = 256;
    // 1) degrees & symmetric normalization
    deg_init_k<<<(N + TB - 1) / TB, TB, 0, stream>>>(dinv, N);
    deg_count_k<<<(E + TB - 1) / TB, TB, 0, stream>>>(col, dinv, E);
    deg_rsqrt_k<<<(N + TB - 1) / TB, TB, 0, stream>>>(dinv, N);
    edge_norm_k<<<(E + TB - 1) / TB, TB, 0, stream>>>(row, col, dinv, norm, E);

    // BN affine precompute (both layers, folds conv biases)
    bn_prep_k<<<1, 128, 0, stream>>>(b1, g1, be1, m1, v1, sc1, sh1, 128);
    bn_prep_k<<<1, 64, 0, stream>>>(b2, g2, be2, m2, v2, sc2, sh2, 64);

    // 2) GEMM1: h1 = x @ W1  (100000x256 @ 256x128), 16x64 tile per wave
    {
        dim3 grid(128 / 64, N / 16);
        gemm_wmma_f32_k<4, 256, 128><<<grid, 32, 0, stream>>>(x, W1, h1);
    }

    // 3) aggregation layer 1: agg1 = D^-1/2 (A+I) D^-1/2 h1
    {
        unsigned t1 = (unsigned)N * 32;
        agg_init_k<32><<<(t1 + TB - 1) / TB, TB, 0, stream>>>(h1, dinv, agg1, t1);
        unsigned t2 = (unsigned)E * 32;
        scatter_edges_k<32><<<(t2 + TB - 1) / TB, TB, 0, stream>>>(h1, row, col, norm, agg1, t2);
    }

    // 4) BN1 + ReLU in place
    {
        unsigned t = (unsigned)N * 128;
        bn_relu_k<128><<<(t + TB - 1) / TB, TB, 0, stream>>>(agg1, sc1, sh1, t);
    }

    // 5) GEMM2: h2 = z1 @ W2  (100000x128 @ 128x64)
    {
        dim3 grid(64 / 64, N / 16);
        gemm_wmma_f32_k<4, 128, 64><<<grid, 32, 0, stream>>>(agg1, W2, h2);
    }

    // 6) aggregation layer 2
    {
        unsigned t1 = (unsigned)N * 16;
        agg_init_k<16><<<(t1 + TB - 1) / TB, TB, 0, stream>>>(h2, dinv, agg2, t1);
        unsigned t2 = (unsigned)E * 16;
        scatter_edges_k<16><<<(t2 + TB - 1) / TB, TB, 0, stream>>>(h2, row, col, norm, agg2, t2);
    }

    // 7) BN2 + ReLU + final linear 64 -> 2
    bn_relu_fc_k<<<(N + TB - 1) / TB, TB, 0, stream>>>(agg2, sc2, sh2, fcW, fcb, out, N);
}